// PQTable_28690381537807
// MI455X (gfx1250) — compile-verified
//
#include <hip/hip_runtime.h>
#include <stdint.h>

// Problem constants (from the reference): S=8 subspaces, K=256 codes,
// B=2, LQ=LK=2048. table: (S,K,K) f32; q_code/k_code: (B,L,S) i32.
#define S_SUB   8
#define K_CODE  256
#define N_B     2
#define LQ_LEN  2048
#define LK_LEN  2048

#define BQ      8        // queries per block  -> 64 gathered rows = 64 KB LDS
#define KT      1024     // keys per block
#define THREADS 256      // 8 wave32 waves

typedef unsigned int v4u __attribute__((ext_vector_type(4)));
typedef int          v8i __attribute__((ext_vector_type(8)));
typedef int          v4i __attribute__((ext_vector_type(4)));

__global__ __launch_bounds__(THREADS)
void pq_score_tdm_kernel(const int*   __restrict__ q_code,
                         const int*   __restrict__ k_code,
                         const float* __restrict__ table,
                         float*       __restrict__ out)
{
    // Gathered table rows: row r = i*S+s holds table[s, q_code[q0+i, s], 0..255].
    // Sole LDS object -> starts at LDS offset 0 (TDM lds_addr is an LDS byte offset).
    __shared__ float rows[BQ * S_SUB * K_CODE];   // 64 KB

    const int  tid   = threadIdx.x;
    const int  wave  = tid >> 5;
    const int  kbase = blockIdx.x * KT;
    const long q0    = (long)blockIdx.y * BQ;       // flattened (b*LQ + lq) query base
    const int  b     = (int)(q0 / LQ_LEN);

    // ------------------------------------------------------------------
    // Stage 1: TDM gather-load. View table as a 2-D tensor of 2048 rows
    // (row index = s*K + code) x 256 f32. Waves 0..3 each issue one
    // TENSOR_LOAD_TO_LDS in gather mode moving 16 rows (16 KB) into LDS.
    // ------------------------------------------------------------------
    if (wave < 4) {
        const int r0 = wave * 16;                   // first flattened row (i*S+s)

        // Pack 16 x 16-bit row indices (block-uniform; force SGPRs).
        unsigned idx[8];
#pragma unroll
        for (int p = 0; p < 8; ++p) {
            const int rA = r0 + 2 * p, rB = rA + 1;
            const int iA = rA >> 3, sA = rA & 7;
            const int iB = rB >> 3, sB = rB & 7;
            const int cA = __builtin_amdgcn_readfirstlane(
                               q_code[(q0 + iA) * S_SUB + sA]);
            const int cB = __builtin_amdgcn_readfirstlane(
                               q_code[(q0 + iB) * S_SUB + sB]);
            const unsigned ia = (unsigned)(sA * K_CODE + cA);
            const unsigned ib = (unsigned)(sB * K_CODE + cB);
            idx[p] = (ia & 0xFFFFu) | (ib << 16);
        }

        const uint64_t ga = (uint64_t)(uintptr_t)table;   // tensor origin

        // D# group 0: count=1 | gather_mode=1 | 16-bit indices; lds byte addr;
        // 57-bit global addr; type=2 ("image") in bits 127:126.
        v4u g0;
        g0.x = 0x80000001u;
        g0.y = (unsigned)(r0 * K_CODE * 4);               // LDS byte offset of row r0
        g0.z = (unsigned)(ga & 0xFFFFFFFFu);
        g0.w = (unsigned)((ga >> 32) & 0x01FFFFFFu) | (2u << 30);

        // D# group 1: data_size=4B; tensor_dim0=256, tensor_dim1=2048;
        // tile_dim0=256, tile_dim1=16 (valid indices); dim0 stride=256.
        v8i g1;
        g1[0] = (int)(2u << 16);                 // workgroup_mask=0, data_size=2 (4B)
        g1[1] = (int)((unsigned)K_CODE << 16);   // tensor_dim0[15:0] in [31:16]
        g1[2] = (int)((unsigned)(S_SUB * K_CODE) << 16); // tensor_dim1[15:0]
        g1[3] = (int)((unsigned)K_CODE << 16);   // tile_dim0 in [127:112]
        g1[4] = 16;                              // tile_dim1 = #valid indices
        g1[5] = K_CODE;                          // tensor_dim0_stride (elements)
        g1[6] = 0;
        g1[7] = 0;

        // D# groups 2/3: the 16 packed 16-bit row indices.
        v4i g2 = { (int)idx[0], (int)idx[1], (int)idx[2], (int)idx[3] };
        v4i g3 = { (int)idx[4], (int)idx[5], (int)idx[6], (int)idx[7] };

        // 6-arg toolchain variant: trailing int32x8 has no D# role for a
        // single descriptor -> zero-filled.
        v8i gz = { 0, 0, 0, 0, 0, 0, 0, 0 };

        __builtin_amdgcn_tensor_load_to_lds(g0, g1, g2, g3, gz, /*cpol=*/0);
        __builtin_amdgcn_s_wait_tensorcnt(0);
    }
    __syncthreads();

    // The TDM engine wrote `rows` behind the compiler's back (descriptor only
    // carries an integer LDS offset). Escape the array's address into an
    // opaque asm with a memory clobber so the LDS loads below are not folded
    // away as "loads from a never-stored global". Reads still go through the
    // original addrspace(3) value -> they lower to ds_load_b32.
    asm volatile("" :: "v"((void*)rows) : "memory");

    // ------------------------------------------------------------------
    // Stage 2: per-key LDS gathers. Lane = key j; 8 ds_load + 8 v_add_f32
    // per output; coalesced non-temporal f32 stores (output is write-once).
    // ------------------------------------------------------------------
    const int*   kc_base  = k_code + (size_t)b * LK_LEN * S_SUB;
    float*       out_base = out + (size_t)q0 * LK_LEN;

#pragma unroll 1
    for (int t = 0; t < KT / THREADS; ++t) {
        const int j = kbase + t * THREADS + tid;

        const v4i c0 = *(const v4i*)(kc_base + (size_t)j * S_SUB);
        const v4i c1 = *(const v4i*)(kc_base + (size_t)j * S_SUB + 4);

        int base[S_SUB];
        base[0] = 0 * K_CODE + c0.x;
        base[1] = 1 * K_CODE + c0.y;
        base[2] = 2 * K_CODE + c0.z;
        base[3] = 3 * K_CODE + c0.w;
        base[4] = 4 * K_CODE + c1.x;
        base[5] = 5 * K_CODE + c1.y;
        base[6] = 6 * K_CODE + c1.z;
        base[7] = 7 * K_CODE + c1.w;

#pragma unroll
        for (int i = 0; i < BQ; ++i) {
            float acc = 0.0f;
#pragma unroll
            for (int s = 0; s < S_SUB; ++s)
                acc += rows[i * (S_SUB * K_CODE) + base[s]];
            __builtin_nontemporal_store(acc, &out_base[(long)i * LK_LEN + j]);
        }
    }
}

extern "C" void kernel_launch(void* const* d_in, const int* in_sizes, int n_in,
                              void* d_out, int out_size, void* d_ws, size_t ws_size,
                              hipStream_t stream) {
    (void)in_sizes; (void)n_in; (void)d_ws; (void)ws_size; (void)out_size;
    const int*   q_code = (const int*)  d_in[0];
    const int*   k_code = (const int*)  d_in[1];
    const float* table  = (const float*)d_in[2];
    float*       out    = (float*)      d_out;

    dim3 grid(LK_LEN / KT, (N_B * LQ_LEN) / BQ);   // 2 x 512 = 1024 blocks
    pq_score_tdm_kernel<<<grid, THREADS, 0, stream>>>(q_code, k_code, table, out);
}